// BaselineMemory_18889266168043
// MI455X (gfx1250) — compile-verified
//
#include <hip/hip_runtime.h>
#include <cstddef>

// CDNA5 / gfx1250: wave32, WMMA f32 16x16x4 path + async global->LDS copies.
typedef __attribute__((ext_vector_type(2))) float v2f;
typedef __attribute__((ext_vector_type(8))) float v8f;
typedef int v4i __attribute__((vector_size(16)));   // matches builtin param type

#define BM 128
#define BN 128
#define BK 16
#define APAD 20    // padded row (floats) for [rows][BK] tiles; 20*l mod 64 distinct for l<16
#define BPAD 136   // padded row (floats) for [BK][BN] tiles; keeps halves on disjoint banks

#if __has_builtin(__builtin_amdgcn_global_load_async_to_lds_b128) && \
    __has_builtin(__builtin_amdgcn_s_wait_asynccnt)
#define USE_ASYNC_LDS 1
#else
#define USE_ASYNC_LDS 0
#endif

#if USE_ASYNC_LDS
typedef __attribute__((address_space(1))) v4i* gv4i_p;
typedef __attribute__((address_space(3))) v4i* lv4i_p;

__device__ __forceinline__ void asyncCopy16(const float* g, float* l) {
    // global_load_async_to_lds_b128: per-lane 16B copy, tracked by ASYNCcnt.
    __builtin_amdgcn_global_load_async_to_lds_b128(
        (gv4i_p)(g), (lv4i_p)(l), /*offset=*/0, /*cpol=*/0);
}
#endif

// ---------------------------------------------------------------------------
// Row L2-normalize: y = x / sqrt(sum(x^2) + 1e-6). One block per row.
// ---------------------------------------------------------------------------
__global__ __launch_bounds__(256)
void rownorm_kernel(const float* __restrict__ X, float* __restrict__ Y, int cols) {
    const int row = blockIdx.x;
    const float* x = X + (size_t)row * cols;
    float* y = Y + (size_t)row * cols;
    __shared__ float red[256];
    float ss = 0.f;
    for (int c = threadIdx.x; c < cols; c += 256) { float v = x[c]; ss += v * v; }
    red[threadIdx.x] = ss; __syncthreads();
    for (int s = 128; s > 0; s >>= 1) {
        if (threadIdx.x < s) red[threadIdx.x] += red[threadIdx.x + s];
        __syncthreads();
    }
    const float scale = 1.0f / sqrtf(red[0] + 1e-6f);
    for (int c = threadIdx.x; c < cols; c += 256) y[c] = x[c] * scale;
}

// ---------------------------------------------------------------------------
// sparsemax over one row of length m (<= 8192), in place.
// Shift invariance => sparsemax(-(1 - S)) == sparsemax(S): work on raw scores.
// Bisection on tau for monotone g(tau) = sum(max(z - tau, 0)) in
// [zmax - 1, zmax]; 48 steps -> fp32-exact tau; then exact renormalization.
// Entire 32KB row lives in LDS (320KB/WGP budget).
// ---------------------------------------------------------------------------
__global__ __launch_bounds__(256)
void sparsemax_kernel(float* __restrict__ S, int m) {
    const int row = blockIdx.x;
    float* z = S + (size_t)row * m;
    __shared__ float zs[8192];
    __shared__ float red[256];
    const int tid = threadIdx.x;

    for (int c = tid; c < m; c += 256) zs[c] = z[c];
    __syncthreads();

    float mx = -__builtin_huge_valf();
    for (int c = tid; c < m; c += 256) mx = fmaxf(mx, zs[c]);
    red[tid] = mx; __syncthreads();
    for (int s = 128; s > 0; s >>= 1) {
        if (tid < s) red[tid] = fmaxf(red[tid], red[tid + s]);
        __syncthreads();
    }
    const float zmax = red[0];
    __syncthreads();

    float lo = zmax - 1.0f, hi = zmax;
    for (int it = 0; it < 48; ++it) {
        const float mid = 0.5f * (lo + hi);
        float s = 0.f;
        for (int c = tid; c < m; c += 256) s += fmaxf(zs[c] - mid, 0.f);
        red[tid] = s; __syncthreads();
        for (int t = 128; t > 0; t >>= 1) {
            if (tid < t) red[tid] += red[tid + t];
            __syncthreads();
        }
        const float tot = red[0];
        __syncthreads();
        if (tot >= 1.0f) lo = mid; else hi = mid;
    }
    const float tau = lo;

    float s = 0.f;
    for (int c = tid; c < m; c += 256) s += fmaxf(zs[c] - tau, 0.f);
    red[tid] = s; __syncthreads();
    for (int t = 128; t > 0; t >>= 1) {
        if (tid < t) red[tid] += red[tid + t];
        __syncthreads();
    }
    const float inv = 1.0f / red[0];
    for (int c = tid; c < m; c += 256) z[c] = fmaxf(zs[c] - tau, 0.f) * inv;
}

// ---------------------------------------------------------------------------
// FP32 WMMA GEMM: C[M,N] = op(A[M,K] x B) (+bias) (+relu)
//   BT=true : B stored row-major [N,K]  (C = A * B^T)  -- TN
//   BT=false: B stored row-major [K,N]  (C = A * B)    -- NN
// Requires M % 128 == 0, K % 16 == 0. N-edge blocks use a synchronous staging
// path; edge-free blocks use double-buffered GLOBAL_LOAD_ASYNC_TO_LDS_B128
// (4 uniform async issues per thread per stage; in-order retire => wait<=4
// after issuing the next stage guarantees the current stage has landed).
// 256 threads = 8 wave32 waves (4 along M x 2 along N); each wave computes a
// 32x64 tile = 2x4 grid of v_wmma_f32_16x16x4_f32 accumulators.
// ---------------------------------------------------------------------------
template<bool BT, bool HASBIAS, bool RELU>
__global__ __launch_bounds__(256)
void gemm_wmma_f32(const float* __restrict__ A, const float* __restrict__ B,
                   const float* __restrict__ bias, float* __restrict__ C,
                   int M, int N, int K) {
    __shared__ float Alds[2][BM * APAD];
    __shared__ float Blds[2][BT ? (BM * APAD) : (BK * BPAD)];

    const int tid   = threadIdx.x;
    const int lane  = tid & 31;
    const int wid   = tid >> 5;
    const int waveM = wid & 3;    // 0..3
    const int waveN = wid >> 2;   // 0..1
    const int l16   = lane & 15;
    const int half  = lane >> 4;  // 0: lanes 0-15, 1: lanes 16-31
    const int blockM = blockIdx.y * BM;
    const int blockN = blockIdx.x * BN;
    const bool edge = (blockN + BN > N);   // only the final GEMM's last N-block

    v8f acc[2][4];
    #pragma unroll
    for (int im = 0; im < 2; ++im)
        #pragma unroll
        for (int jn = 0; jn < 4; ++jn) acc[im][jn] = (v8f)0.f;

    // ---- compute 4 WMMA k-steps over one staged BK tile ----
    auto computeTile = [&](int buf) {
        #pragma unroll
        for (int kk = 0; kk < BK; kk += 4) {
            v2f a[2], b[4];
            // A frag 16x4: lane<16 holds {K=kk,kk+1}, lane>=16 holds {kk+2,kk+3}
            #pragma unroll
            for (int im = 0; im < 2; ++im) {
                const float* p = &Alds[buf][(waveM * 32 + im * 16 + l16) * APAD + kk + half * 2];
                a[im].x = p[0]; a[im].y = p[1];
            }
            // B frag 4x16: VGPR j = row K=kk+j (lanes 0-15) / K=kk+2+j (lanes 16-31)
            #pragma unroll
            for (int jn = 0; jn < 4; ++jn) {
                const int ncol = waveN * 64 + jn * 16 + l16;
                if (BT) {   // B^T tile has the same per-lane layout as an A tile
                    const float* p = &Blds[buf][ncol * APAD + kk + half * 2];
                    b[jn].x = p[0]; b[jn].y = p[1];
                } else {
                    b[jn].x = Blds[buf][(kk + half * 2 + 0) * BPAD + ncol];
                    b[jn].y = Blds[buf][(kk + half * 2 + 1) * BPAD + ncol];
                }
            }
            #pragma unroll
            for (int im = 0; im < 2; ++im)
                #pragma unroll
                for (int jn = 0; jn < 4; ++jn)
                    acc[im][jn] = __builtin_amdgcn_wmma_f32_16x16x4_f32(
                        false, a[im], false, b[jn], (short)0, acc[im][jn], false, false);
        }
    };

#if USE_ASYNC_LDS
    // ---- async staging: exactly 4 x b128 per thread per stage, no guards ----
    auto issueAsync = [&](int buf, int kt) {
        #pragma unroll
        for (int it = 0; it < 2; ++it) {
            const int i = tid + it * 256;
            { const int r = i >> 2, c4 = i & 3;
              asyncCopy16(A + (size_t)(blockM + r) * K + kt + c4 * 4,
                          &Alds[buf][r * APAD + c4 * 4]); }
            if constexpr (BT) {
                const int r = i >> 2, c4 = i & 3;
                asyncCopy16(B + (size_t)(blockN + r) * K + kt + c4 * 4,
                            &Blds[buf][r * APAD + c4 * 4]);
            } else {
                const int r = i >> 5, c4 = i & 31;
                asyncCopy16(B + (size_t)(kt + r) * N + blockN + c4 * 4,
                            &Blds[buf][r * BPAD + c4 * 4]);
            }
        }
    };
#endif
    // ---- synchronous staging (edge blocks / builtin-less toolchain) ----
    auto stageSync = [&](int buf, int kt) {
        #pragma unroll
        for (int it = 0; it < 2; ++it) {
            const int i = tid + it * 256;
            { const int r = i >> 2, c4 = i & 3;
              float4 v = *(const float4*)(A + (size_t)(blockM + r) * K + kt + c4 * 4);
              *(float4*)(&Alds[buf][r * APAD + c4 * 4]) = v; }
            if constexpr (BT) {
                const int r = i >> 2, c4 = i & 3;
                float4 v = {0.f, 0.f, 0.f, 0.f};
                if (blockN + r < N)
                    v = *(const float4*)(B + (size_t)(blockN + r) * K + kt + c4 * 4);
                *(float4*)(&Blds[buf][r * APAD + c4 * 4]) = v;
            } else {
                const int r = i >> 5, c4 = i & 31;
                const int gcol = blockN + c4 * 4;
                const float* bp = B + (size_t)(kt + r) * N + gcol;
                float4 v = {0.f, 0.f, 0.f, 0.f};
                if (gcol + 3 < N) {
                    v = *(const float4*)bp;
                } else {
                    float t[4] = {0.f, 0.f, 0.f, 0.f};
                    for (int j = 0; j < 4; ++j)
                        if (gcol + j < N) t[j] = bp[j];
                    v = {t[0], t[1], t[2], t[3]};
                }
                *(float4*)(&Blds[buf][r * BPAD + c4 * 4]) = v;
            }
        }
    };

    int buf = 0;
#if USE_ASYNC_LDS
    if (!edge) {
        issueAsync(0, 0);
        for (int kt = 0; kt < K; kt += BK) {
            if (kt + BK < K) {
                issueAsync(buf ^ 1, kt + BK);
                __builtin_amdgcn_s_wait_asynccnt(4);  // 4 newest (next stage) may remain
            } else {
                __builtin_amdgcn_s_wait_asynccnt(0);
            }
            __syncthreads();          // stage `buf` visible to all waves
            computeTile(buf);
            __syncthreads();          // all reads of `buf` done before it is refilled
            buf ^= 1;
        }
    } else
#endif
    {
        stageSync(0, 0);
        for (int kt = 0; kt < K; kt += BK) {
            __syncthreads();
            if (kt + BK < K) stageSync(buf ^ 1, kt + BK);
            computeTile(buf);
            __syncthreads();
            buf ^= 1;
        }
    }

    // Epilogue: C/D layout — VGPR i: lanes 0-15 -> M=i, lanes 16-31 -> M=i+8
    #pragma unroll
    for (int im = 0; im < 2; ++im) {
        #pragma unroll
        for (int jn = 0; jn < 4; ++jn) {
            const int col = blockN + waveN * 64 + jn * 16 + l16;
            if (col >= N) continue;
            const int rbase = blockM + waveM * 32 + im * 16 + half * 8;
            const float bv = HASBIAS ? bias[col] : 0.f;
            #pragma unroll
            for (int i = 0; i < 8; ++i) {
                float v = acc[im][jn][i] + bv;
                if (RELU) v = fmaxf(v, 0.f);
                C[(size_t)(rbase + i) * N + col] = v;
            }
        }
    }
}

// ---------------------------------------------------------------------------
// Orchestration. ws layout (floats):
//   Xn[b*d] | Yn[m*d] | S[b*m] | Mv[b*d] | H[b*h]   (~128 MB total)
// ---------------------------------------------------------------------------
extern "C" void kernel_launch(void* const* d_in, const int* in_sizes, int n_in,
                              void* d_out, int out_size, void* d_ws, size_t ws_size,
                              hipStream_t stream) {
    const float* enc = (const float*)d_in[0];  // [b, d]
    const float* mem = (const float*)d_in[1];  // [m, d]
    const float* W1  = (const float*)d_in[2];  // [d, 2d]
    const float* b1  = (const float*)d_in[3];  // [2d]
    const float* W2  = (const float*)d_in[4];  // [2d, o]
    const float* b2  = (const float*)d_in[5];  // [o]
    float* out = (float*)d_out;

    const int h = in_sizes[3];            // 2d = 2048
    const int d = h / 2;                  // 1024
    const int b = in_sizes[0] / d;        // 2048
    const int m = in_sizes[1] / d;        // 8192
    const int o = in_sizes[5];            // 1000

    float* ws = (float*)d_ws;
    float* Xn = ws;
    float* Yn = Xn + (size_t)b * d;
    float* S  = Yn + (size_t)m * d;
    float* Mv = S  + (size_t)b * m;
    float* H  = Mv + (size_t)b * d;

    // 1) row-normalize both operands of the cosine kernel
    rownorm_kernel<<<b, 256, 0, stream>>>(enc, Xn, d);
    rownorm_kernel<<<m, 256, 0, stream>>>(mem, Yn, d);

    // 2) S = Xn @ Yn^T   (sparsemax(-dist) == sparsemax(S) by shift invariance)
    dim3 g1(m / BN, b / BM);
    gemm_wmma_f32<true, false, false><<<g1, 256, 0, stream>>>(Xn, Yn, nullptr, S, b, m, d);

    // 3) sparsemax rows in place
    sparsemax_kernel<<<b, 256, 0, stream>>>(S, m);

    // 4) Mv = S @ memory_set
    dim3 g2(d / BN, b / BM);
    gemm_wmma_f32<false, false, false><<<g2, 256, 0, stream>>>(S, mem, nullptr, Mv, b, d, m);

    // 5) H = relu(Mv @ W1 + b1)
    dim3 g3(h / BN, b / BM);
    gemm_wmma_f32<false, true, true><<<g3, 256, 0, stream>>>(Mv, W1, b1, H, b, h, d);

    // 6) out = H @ W2 + b2  (N = 1000, edge block uses the sync path)
    dim3 g4((o + BN - 1) / BN, b / BM);
    gemm_wmma_f32<false, true, false><<<g4, 256, 0, stream>>>(H, W2, b2, out, b, o, h);
}